// DepthwiseConvolutionND_90898687853198
// MI455X (gfx1250) — compile-verified
//
#include <hip/hip_runtime.h>
#include <hip/hip_bf16.h>
#include <stdint.h>

// Depthwise 3x3 SAME conv, shared single-channel kernel, multiplier 2.
// x: [16,224,224,96] f32 NHWC; kernel: [3,3,1,2]; bias: [192]; out: [16,224,224,192].
// Bandwidth-bound (AI ~3 FLOP/B) -> TDM async tile load to LDS + VALU FMA compute,
// coalesced float2 output stores. 14x8 spatial tile (16x10 halo, 61.4 KB LDS).

#define B_   16
#define H_   224
#define W_   224
#define C_   96
#define TH_  14
#define TW_  8
#define HALO_H (TH_ + 2)              // 16
#define HALO_W (TW_ + 2)              // 10
#define PITCH  (HALO_W * C_)          // 960 floats per LDS tile row
#define TILE_ELEMS (HALO_H * PITCH)   // 15360 floats = 61440 B

typedef unsigned int       u32;
typedef unsigned long long u64;
typedef u32 v4u __attribute__((ext_vector_type(4)));
typedef int v8i __attribute__((ext_vector_type(8)));
typedef int v4i __attribute__((ext_vector_type(4)));

#if defined(__has_builtin)
#if __has_builtin(__builtin_amdgcn_tensor_load_to_lds)
#define HAVE_TDM 1
#endif
#endif

#ifdef HAVE_TDM
// One TDM row load: 1-row 2D tile, dim0 flattened over (w,c) which is contiguous in NHWC.
__device__ __forceinline__ void tdm_load_row(u32 lds_byte_addr, u64 gbyte_addr,
                                             u32 tensor_d0, u32 tile_d0, u64 stride0) {
  v4u g0;
  g0.x = 1u;                                   // count=1 (valid user descriptor), gather off
  g0.y = lds_byte_addr;                        // bits 63:32  lds_addr
  g0.z = (u32)(gbyte_addr & 0xFFFFFFFFull);    // bits 95:64  global_addr[31:0]
  g0.w = (u32)((gbyte_addr >> 32) & 0x01FFFFFFull) | (2u << 30); // addr[56:32] | type=2

  v8i g1;
  g1[0] = (int)(2u << 16);                     // workgroup_mask=0, data_size=2 (4 bytes)
  g1[1] = (int)((tensor_d0 & 0xFFFFu) << 16);  // tensor_dim0[15:0]  (bits 63:48)
  g1[2] = (int)((tensor_d0 >> 16) | (1u << 16));       // tensor_dim0[31:16] | tensor_dim1=1
  g1[3] = (int)((tile_d0 & 0xFFFFu) << 16);    // tensor_dim1 hi = 0 | tile_dim0
  g1[4] = (int)1;                              // tile_dim1=1, tile_dim2=0
  g1[5] = (int)(stride0 & 0xFFFFFFFFull);      // tensor_dim0_stride[31:0]
  g1[6] = (int)((stride0 >> 32) & 0xFFFFull);  // stride[47:32] | dim1_stride lo = 0
  g1[7] = 0;

  v4i z4 = (v4i){0, 0, 0, 0};
  v8i z8 = (v8i){0, 0, 0, 0, 0, 0, 0, 0};
  // clang-23 / amdgpu-toolchain 6-arg form:
  // (uint32x4 g0, int32x8 g1, int32x4 g2, int32x4 g3, int32x8 ext, i32 cpol)
  __builtin_amdgcn_tensor_load_to_lds(g0, g1, z4, z4, z8, 0);
}
#endif

__global__ __launch_bounds__(256)
void DepthwiseConvolutionND_90898687853198_kernel(const float* __restrict__ x,
                                                  const float* __restrict__ kernel,
                                                  const float* __restrict__ bias,
                                                  float* __restrict__ out) {
  __shared__ float tile[TILE_ELEMS];

  const int tid = threadIdx.x;
  const int b  = blockIdx.z;
  const int h0 = blockIdx.y * TH_;
  const int w0 = blockIdx.x * TW_;

  // Zeros are only needed for halo cells TDM never writes: left column
  // (blockIdx.x==0) and skipped top/bottom rows. Right-edge zeros come free
  // from TDM tensor_dim0 OOB semantics. Interior blocks skip the memset.
  const bool need_zero = (blockIdx.x == 0) || (blockIdx.y == 0) ||
                         (blockIdx.y == gridDim.y - 1);
  if (need_zero) {
    for (int i = tid; i < TILE_ELEMS; i += 256) tile[i] = 0.0f;
  }
  __syncthreads();

  const int gh0 = h0 - 1;
  const int gw0 = w0 - 1;
  const int ws = gw0 < 0 ? 0 : gw0;      // clamped left column
  const int col_skip = ws - gw0;         // 0 or 1

#ifdef HAVE_TDM
  if (tid < 32) {                        // wave 0 issues the DMA descriptors
    const u32 ldsbase = (u32)(uintptr_t)(&tile[0]);   // low 32 bits = LDS byte offset
    const u32 tensor_d0 = (u32)(W_ - ws) * C_;        // right-edge OOB -> zeros
    const u32 tile_d0   = (u32)(HALO_W - col_skip) * C_;
    const u64 stride0   = (u64)tensor_d0;
    for (int r = 0; r < HALO_H; ++r) {
      const int gh = gh0 + r;
      if (gh < 0 || gh >= H_) continue;  // top/bottom halo rows stay zero
      const u32 lds_row = ldsbase + (u32)((r * PITCH + col_skip * C_) * 4);
      const u64 gaddr = (u64)(uintptr_t)x +
                        ((u64)((b * H_ + gh) * W_ + ws) * C_) * 4ull;
      tdm_load_row(lds_row, gaddr, tensor_d0, tile_d0, stride0);
    }
    __builtin_amdgcn_s_wait_tensorcnt(0);
  }
#else
  // Fallback: coalesced global->LDS copy (lanes differ in flattened (w,c): contiguous).
  for (int r = 0; r < HALO_H; ++r) {
    const int gh = gh0 + r;
    if (gh < 0 || gh >= H_) continue;
    const int n = (HALO_W - col_skip) * C_;
    const float* src = x + ((size_t)((b * H_ + gh) * W_ + ws)) * C_;
    const int lim = (W_ - ws) * C_;
    float* dst = &tile[r * PITCH + col_skip * C_];
    for (int i = tid; i < n; i += 256) dst[i] = (i < lim) ? src[i] : 0.0f;
  }
#endif
  __syncthreads();

  // Shared 3x3 kernel taps, multiplier 2: k[t][m], t = kh*3+kw.
  float k0[9], k1[9];
#pragma unroll
  for (int t = 0; t < 9; ++t) {
    const float2 kv = ((const float2*)kernel)[t];
    k0[t] = kv.x;
    k1[t] = kv.y;
  }

  const float2* bias2 = (const float2*)bias;
  float2* out2 = (float2*)out;

  // 14*8*96 = 10752 output (pixel,channel) pairs; 256 threads -> 42 iters each.
  for (int idx = tid; idx < TH_ * TW_ * C_; idx += 256) {
    const int c = idx % C_;
    const int rem = idx / C_;
    const int w = rem % TW_;
    const int h = rem / TW_;

    const float2 bv = bias2[c];
    float a0 = bv.x, a1 = bv.y;

#pragma unroll
    for (int kh = 0; kh < 3; ++kh) {
#pragma unroll
      for (int kw = 0; kw < 3; ++kw) {
        const float v = tile[(h + kh) * PITCH + (w + kw) * C_ + c];
        const int t = kh * 3 + kw;
        a0 = fmaf(v, k0[t], a0);
        a1 = fmaf(v, k1[t], a1);
      }
    }

    // out channel ordering c*2+m -> one float2 store at channel pair c.
    const size_t pix = (size_t)(b * H_ + (h0 + h)) * W_ + (w0 + w);
    out2[pix * C_ + c] = make_float2(a0, a1);
  }
}

extern "C" void kernel_launch(void* const* d_in, const int* in_sizes, int n_in,
                              void* d_out, int out_size, void* d_ws, size_t ws_size,
                              hipStream_t stream) {
  const float* x      = (const float*)d_in[0];
  const float* kernel = (const float*)d_in[1];
  const float* bias   = (const float*)d_in[2];
  float* out          = (float*)d_out;

  dim3 grid(W_ / TW_, H_ / TH_, B_);   // (28, 16, 16)
  dim3 block(256, 1, 1);
  DepthwiseConvolutionND_90898687853198_kernel<<<grid, block, 0, stream>>>(x, kernel, bias, out);
}